// Encoder_86431921865133
// MI455X (gfx1250) — compile-verified
//
#include <hip/hip_runtime.h>
#include <math.h>

// ---------------------------------------------------------------------------
// Graph-ODE GAT encoder for MI455X (gfx1250, wave32).
// GEMMs run on V_WMMA_F32_16X16X4_F32 (full fp32 fidelity; workload is
// latency/serialization bound, not FLOP bound, so no precision downcast).
// Elementwise/staging traffic vectorized to b128; weights prefetched via
// global_prefetch_b8.
// ---------------------------------------------------------------------------

namespace {
constexpr int B = 8, SEQ = 12, N = 1000, IN_DIM = 2;
constexpr int HID = 128, HEADS = 8, DH = 16, E = 16000, INTERVAL = 4;
constexpr float SCALE = 1.0f, SLOPE = 0.2f;
constexpr int BN    = B * N;          // 8000 rows
constexpr int STATE = BN * HID;       // 1,024,000 floats (divisible by 4)
constexpr int BNH   = BN * HEADS;     // 64,000 (divisible by 4)
constexpr int BEH   = B * E * HEADS;  // 1,024,000 (== STATE)

typedef __attribute__((ext_vector_type(2))) float v2f;
typedef __attribute__((ext_vector_type(4))) float v4f;
typedef __attribute__((ext_vector_type(8))) float v8f;
} // namespace

// ---------------------------------------------------------------------------
// WMMA GEMM: C[M x 128] = op( A1@W1 (+ A2@W2) + b1 + b2 )
// One wave (32 lanes) per 16-row tile; 8 column tiles (16 cols each = 1 head);
// K = 128 as 32 steps of 16x16x4 fp32 WMMA. A tiles staged in LDS with b128.
// All call sites use M % 16 == 0 (M = 8000).
//
// fp32 WMMA fragment layouts (ISA 7.12.2):
//   A (16x4): a[j] = A[M = lane&15][K = j + 2*(lane>>4)]
//   B (4x16): b[j] = B[K = j + 2*(lane>>4)][N = lane&15]
//   C (16x16): c[r] = C[M = r + 8*(lane>>4)][N = lane&15]
// ---------------------------------------------------------------------------
__global__ __launch_bounds__(32)
void gemm16_wmma(const float* __restrict__ A1, const float* __restrict__ W1,
                 const float* __restrict__ A2, const float* __restrict__ W2,
                 const float* __restrict__ b1, const float* __restrict__ b2,
                 float* __restrict__ C, int M, int do_tanh)
{
    __shared__ float As1[16 * HID];
    __shared__ float As2[16 * HID];
    const int lane = threadIdx.x;          // 0..31, EXEC all ones (no divergence)
    const int row0 = blockIdx.x * 16;

    // Warm the WGP cache with the weight matrices (emits global_prefetch_b8;
    // one probe per 32 floats = one per 128B line).
    for (int i = lane * 32; i < HID * HID; i += 32 * 32) {
        __builtin_prefetch(&W1[i], 0, 0);
        if (W2) __builtin_prefetch(&W2[i], 0, 0);
    }

    // Stage A tile(s) into LDS: 512 b128 transactions, fully coalesced.
    {
        const v4f* a1v = (const v4f*)(A1 + (size_t)row0 * HID);
        v4f* s1v = (v4f*)As1;
        for (int i = lane; i < 16 * HID / 4; i += 32) s1v[i] = a1v[i];
        if (A2) {
            const v4f* a2v = (const v4f*)(A2 + (size_t)row0 * HID);
            v4f* s2v = (v4f*)As2;
            for (int i = lane; i < 16 * HID / 4; i += 32) s2v[i] = a2v[i];
        }
    }
    __syncthreads();

    const int hi = lane >> 4;   // half-wave select (K pairs / M+8)
    const int lo = lane & 15;

    for (int ct = 0; ct < HID / 16; ++ct) {      // 8 column tiles (= heads)
        v8f acc = {};
        const int col = ct * 16 + lo;
        #pragma unroll
        for (int kk = 0; kk < HID / 4; ++kk) {
            const int k = 4 * kk + 2 * hi;
            v2f a, b;
            a.x = As1[lo * HID + k];
            a.y = As1[lo * HID + k + 1];
            b.x = W1[(k    ) * HID + col];
            b.y = W1[(k + 1) * HID + col];
            acc = __builtin_amdgcn_wmma_f32_16x16x4_f32(
                false, a, false, b, (short)0, acc, false, false);
        }
        if (A2) {                                 // fused second GEMM (y@w_W + h@w_U)
            #pragma unroll
            for (int kk = 0; kk < HID / 4; ++kk) {
                const int k = 4 * kk + 2 * hi;
                v2f a, b;
                a.x = As2[lo * HID + k];
                a.y = As2[lo * HID + k + 1];
                b.x = W2[(k    ) * HID + col];
                b.y = W2[(k + 1) * HID + col];
                acc = __builtin_amdgcn_wmma_f32_16x16x4_f32(
                    false, a, false, b, (short)0, acc, false, false);
            }
        }
        float badd = 0.0f;
        if (b1) badd += b1[col];
        if (b2) badd += b2[col];
        #pragma unroll
        for (int r = 0; r < 8; ++r) {
            const int row = row0 + r + 8 * hi;
            if (row < M) {
                float v = acc[r] + badd;
                if (do_tanh) v = tanhf(v);
                C[row * HID + col] = v;
            }
        }
    }
}

// ---------------------------------------------------------------------------
// h[:, sidx] = inputs[:, sidx] @ w_in + b_in   (K = 2, not worth WMMA)
// ---------------------------------------------------------------------------
__global__ void input_proj(const float* __restrict__ in, const float* __restrict__ w_in,
                           const float* __restrict__ b_in, float* __restrict__ hout, int sidx)
{
    const int idx = blockIdx.x * blockDim.x + threadIdx.x;
    if (idx >= STATE) return;
    const int c  = idx & 127;
    const int bn = idx >> 7;
    const int b  = bn / N, n = bn % N;
    const float* ip = in + ((size_t)(b * SEQ + sidx) * N + n) * IN_DIM;
    hout[idx] = ip[0] * w_in[c] + ip[1] * w_in[HID + c] + b_in[c];
}

// el/er[b,n,h] = <hproj[b,n,h,:], a_l/a_r[h,:]>
__global__ void attn_logits(const float* __restrict__ hproj,
                            const float* __restrict__ a_l, const float* __restrict__ a_r,
                            float* __restrict__ el, float* __restrict__ er)
{
    const int idx = blockIdx.x * blockDim.x + threadIdx.x;
    if (idx >= BNH) return;
    const int h   = idx % HEADS;
    const int row = idx / HEADS;                // b*N+n
    const v4f* hp = (const v4f*)(hproj + (size_t)row * HID + h * DH);
    const v4f* al = (const v4f*)(a_l + h * DH);
    const v4f* ar = (const v4f*)(a_r + h * DH);
    float sl = 0.f, sr = 0.f;
    #pragma unroll
    for (int q = 0; q < DH / 4; ++q) {
        const v4f hv = hp[q], alv = al[q], arv = ar[q];
        sl += hv.x * alv.x + hv.y * alv.y + hv.z * alv.z + hv.w * alv.w;
        sr += hv.x * arv.x + hv.y * arv.y + hv.z * arv.z + hv.w * arv.w;
    }
    el[idx] = sl;
    er[idx] = sr;
}

// Vectorized fills/copies: n4 = element count / 4 (all sizes divisible by 4).
__global__ void fill_f32v4(v4f* __restrict__ p, float v, int n4)
{
    const int i = blockIdx.x * blockDim.x + threadIdx.x;
    if (i < n4) { v4f q = {v, v, v, v}; p[i] = q; }
}

__global__ void copy_f32v4(v4f* __restrict__ dst, const v4f* __restrict__ src, int n4)
{
    const int i = blockIdx.x * blockDim.x + threadIdx.x;
    if (i < n4) dst[i] = src[i];
}

__device__ __forceinline__ void atomicMaxF(float* addr, float val)
{
    unsigned* ia = (unsigned*)addr;
    unsigned old = __float_as_uint(*addr);
    while (__uint_as_float(old) < val) {
        const unsigned assumed = old;
        old = atomicCAS(ia, assumed, __float_as_uint(val));
        if (old == assumed) break;
    }
}

// score = leaky_relu(el[src] + er[dst]); segment-max into m[dst]
__global__ void edge_scores_max(const float* __restrict__ el, const float* __restrict__ er,
                                const int* __restrict__ src, const int* __restrict__ dst,
                                float* __restrict__ ebuf, float* __restrict__ m)
{
    const int idx = blockIdx.x * blockDim.x + threadIdx.x;
    if (idx >= BEH) return;
    const int h = idx % HEADS;
    const int e = (idx / HEADS) % E;
    const int b = idx / (HEADS * E);
    const int sn = src[e], dn = dst[e];
    float v = el[(b * N + sn) * HEADS + h] + er[(b * N + dn) * HEADS + h];
    v = (v > 0.f) ? v : SLOPE * v;
    ebuf[idx] = v;
    atomicMaxF(&m[(b * N + dn) * HEADS + h], v);
}

// ex = exp(score - m[dst]); segment-sum into s[dst]
__global__ void edge_exp_sum(const int* __restrict__ dst, const float* __restrict__ m,
                             float* __restrict__ ebuf, float* __restrict__ s)
{
    const int idx = blockIdx.x * blockDim.x + threadIdx.x;
    if (idx >= BEH) return;
    const int h = idx % HEADS;
    const int e = (idx / HEADS) % E;
    const int b = idx / (HEADS * E);
    const int dn = dst[e];
    const float ex = __expf(ebuf[idx] - m[(b * N + dn) * HEADS + h]);
    ebuf[idx] = ex;
    atomicAdd(&s[(b * N + dn) * HEADS + h], ex);
}

// agg[dst] += alpha * hproj[src]  (16-wide per head, float atomics)
__global__ void edge_aggregate(const int* __restrict__ src, const int* __restrict__ dst,
                               const float* __restrict__ ebuf, const float* __restrict__ ssum,
                               const float* __restrict__ hproj, float* __restrict__ agg)
{
    const int idx = blockIdx.x * blockDim.x + threadIdx.x;
    if (idx >= BEH) return;
    const int h = idx % HEADS;
    const int e = (idx / HEADS) % E;
    const int b = idx / (HEADS * E);
    const int sn = src[e], dn = dst[e];
    const float alpha = ebuf[idx] / (ssum[(b * N + dn) * HEADS + h] + 1e-16f);
    const float* hp = hproj + (size_t)(b * N + sn) * HID + h * DH;
    float*       ag = agg   + (size_t)(b * N + dn) * HID + h * DH;
    #pragma unroll
    for (int d = 0; d < DH; ++d) atomicAdd(&ag[d], hp[d] * alpha);
}

// out = y + a*k  (vectorized)
__global__ void axpy_v4(v4f* __restrict__ out, const v4f* __restrict__ y,
                        const v4f* __restrict__ k, float a, int n4)
{
    const int i = blockIdx.x * blockDim.x + threadIdx.x;
    if (i < n4) {
        const v4f yv = y[i], kv = k[i];
        v4f o;
        o.x = yv.x + a * kv.x; o.y = yv.y + a * kv.y;
        o.z = yv.z + a * kv.z; o.w = yv.w + a * kv.w;
        out[i] = o;
    }
}

// ksum = w*k (overwrite) or ksum += w*k  (vectorized)
__global__ void kacc_v4(v4f* __restrict__ ks, const v4f* __restrict__ k,
                        float w, int overwrite, int n4)
{
    const int i = blockIdx.x * blockDim.x + threadIdx.x;
    if (i < n4) {
        const v4f kv = k[i];
        v4f o;
        if (overwrite) {
            o.x = w * kv.x; o.y = w * kv.y; o.z = w * kv.z; o.w = w * kv.w;
        } else {
            const v4f sv = ks[i];
            o.x = sv.x + w * kv.x; o.y = sv.y + w * kv.y;
            o.z = sv.z + w * kv.z; o.w = sv.w + w * kv.w;
        }
        ks[i] = o;
    }
}

// LayerNorm over last dim (128): one wave32 per row, __shfl_xor reductions.
__global__ __launch_bounds__(256)
void layernorm(float* __restrict__ y, int rows)
{
    const int gw   = (blockIdx.x * blockDim.x + threadIdx.x) >> 5;
    const int lane = threadIdx.x & 31;
    if (gw >= rows) return;
    float* row = y + (size_t)gw * HID;
    float v[4];
    float sum = 0.f;
    #pragma unroll
    for (int j = 0; j < 4; ++j) { v[j] = row[lane + 32 * j]; sum += v[j]; }
    #pragma unroll
    for (int o = 16; o > 0; o >>= 1) sum += __shfl_xor(sum, o, 32);
    const float mu = sum * (1.0f / HID);
    float var = 0.f;
    #pragma unroll
    for (int j = 0; j < 4; ++j) { const float d = v[j] - mu; var += d * d; }
    #pragma unroll
    for (int o = 16; o > 0; o >>= 1) var += __shfl_xor(var, o, 32);
    const float inv = rsqrtf(var * (1.0f / HID) + 1e-5f);
    #pragma unroll
    for (int j = 0; j < 4; ++j) row[lane + 32 * j] = (v[j] - mu) * inv;
}

// ---------------------------------------------------------------------------
extern "C" void kernel_launch(void* const* d_in, const int* in_sizes, int n_in,
                              void* d_out, int out_size, void* d_ws, size_t ws_size,
                              hipStream_t stream)
{
    (void)in_sizes; (void)n_in; (void)out_size; (void)ws_size;

    const float* inputs = (const float*)d_in[0];
    const int*   src    = (const int*)d_in[1];
    const int*   dst    = (const int*)d_in[2];
    const float* w_in   = (const float*)d_in[3];
    const float* b_in   = (const float*)d_in[4];
    const float* w_gat  = (const float*)d_in[5];
    const float* a_l    = (const float*)d_in[6];
    const float* a_r    = (const float*)d_in[7];
    const float* w_W    = (const float*)d_in[8];
    const float* b_W    = (const float*)d_in[9];
    const float* w_U    = (const float*)d_in[10];
    const float* b_U    = (const float*)d_in[11];
    const float* w_o1   = (const float*)d_in[12];
    const float* b_o1   = (const float*)d_in[13];
    const float* w_o2   = (const float*)d_in[14];
    const float* b_o2   = (const float*)d_in[15];
    float* out = (float*)d_out;

    // Workspace layout (floats): ~30 MB total.
    float* ws    = (float*)d_ws;
    float* y     = ws + (size_t)0 * STATE;   // ODE state
    float* ytmp  = ws + (size_t)1 * STATE;   // RK4 stage input
    float* kcur  = ws + (size_t)2 * STATE;   // current k (GAT output)
    float* ksum  = ws + (size_t)3 * STATE;   // k1 + 2k2 + 2k3 + k4
    float* hproj = ws + (size_t)4 * STATE;   // y @ w_gat  (also reused as temp)
    float* hidx  = ws + (size_t)5 * STATE;   // h[:, idx]
    float* ebuf  = ws + (size_t)6 * STATE;   // edge scores/exps (BEH == STATE)
    float* el    = ws + (size_t)7 * STATE;
    float* er    = el + BNH;
    float* mbuf  = er + BNH;
    float* sbuf  = mbuf + BNH;

    const dim3 blk(256);
    auto nb = [](int n, int t) { return dim3((unsigned)((n + t - 1) / t)); };
    const float dt = SCALE / (float)INTERVAL;
    const int STATE4 = STATE / 4, BNH4 = BNH / 4;

    // f(x) = GAT(x): projection (WMMA) -> logits -> edge softmax -> aggregate
    auto feval = [&](const float* x, float* k) {
        gemm16_wmma<<<BN / 16, 32, 0, stream>>>(x, w_gat, nullptr, nullptr,
                                                nullptr, nullptr, hproj, BN, 0);
        attn_logits<<<nb(BNH, 256), blk, 0, stream>>>(hproj, a_l, a_r, el, er);
        fill_f32v4<<<nb(BNH4, 256), blk, 0, stream>>>((v4f*)mbuf, -3.402823466e38f, BNH4);
        fill_f32v4<<<nb(BNH4, 256), blk, 0, stream>>>((v4f*)sbuf, 0.0f, BNH4);
        fill_f32v4<<<nb(STATE4, 256), blk, 0, stream>>>((v4f*)k, 0.0f, STATE4);
        edge_scores_max<<<nb(BEH, 256), blk, 0, stream>>>(el, er, src, dst, ebuf, mbuf);
        edge_exp_sum<<<nb(BEH, 256), blk, 0, stream>>>(dst, mbuf, ebuf, sbuf);
        edge_aggregate<<<nb(BEH, 256), blk, 0, stream>>>(src, dst, ebuf, sbuf, hproj, k);
    };

    // ret = h[:, 0]
    input_proj<<<nb(STATE, 256), blk, 0, stream>>>(inputs, w_in, b_in, y, 0);

    for (int idx = 0; idx < SEQ; ++idx) {
        // RK4 over INTERVAL steps, dt = SCALE/INTERVAL
        for (int step = 0; step < INTERVAL; ++step) {
            feval(y, kcur);                                                        // k1
            kacc_v4<<<nb(STATE4, 256), blk, 0, stream>>>((v4f*)ksum, (const v4f*)kcur, 1.0f, 1, STATE4);
            axpy_v4<<<nb(STATE4, 256), blk, 0, stream>>>((v4f*)ytmp, (const v4f*)y, (const v4f*)kcur, 0.5f * dt, STATE4);
            feval(ytmp, kcur);                                                     // k2
            kacc_v4<<<nb(STATE4, 256), blk, 0, stream>>>((v4f*)ksum, (const v4f*)kcur, 2.0f, 0, STATE4);
            axpy_v4<<<nb(STATE4, 256), blk, 0, stream>>>((v4f*)ytmp, (const v4f*)y, (const v4f*)kcur, 0.5f * dt, STATE4);
            feval(ytmp, kcur);                                                     // k3
            kacc_v4<<<nb(STATE4, 256), blk, 0, stream>>>((v4f*)ksum, (const v4f*)kcur, 2.0f, 0, STATE4);
            axpy_v4<<<nb(STATE4, 256), blk, 0, stream>>>((v4f*)ytmp, (const v4f*)y, (const v4f*)kcur, dt, STATE4);
            feval(ytmp, kcur);                                                     // k4
            kacc_v4<<<nb(STATE4, 256), blk, 0, stream>>>((v4f*)ksum, (const v4f*)kcur, 1.0f, 0, STATE4);
            axpy_v4<<<nb(STATE4, 256), blk, 0, stream>>>((v4f*)y, (const v4f*)y, (const v4f*)ksum, dt / 6.0f, STATE4);
        }
        if (idx != 0) {
            // ret = tanh(ret@w_W + b_W + h[:,idx]@w_U + b_U)  (fused dual WMMA GEMM)
            input_proj<<<nb(STATE, 256), blk, 0, stream>>>(inputs, w_in, b_in, hidx, idx);
            gemm16_wmma<<<BN / 16, 32, 0, stream>>>(y, w_W, hidx, w_U,
                                                    b_W, b_U, hproj, BN, 1);
            copy_f32v4<<<nb(STATE4, 256), blk, 0, stream>>>((v4f*)y, (const v4f*)hproj, STATE4);
        }
        layernorm<<<nb(BN * 32, 256), blk, 0, stream>>>(y, BN);
    }

    // out = tanh(ret@w_o1 + b_o1) @ w_o2 + b_o2
    gemm16_wmma<<<BN / 16, 32, 0, stream>>>(y, w_o1, nullptr, nullptr,
                                            b_o1, nullptr, hproj, BN, 1);
    gemm16_wmma<<<BN / 16, 32, 0, stream>>>(hproj, w_o2, nullptr, nullptr,
                                            b_o2, nullptr, out, BN, 0);
}